// ChemicalDevelopment_34883724378574
// MI455X (gfx1250) — compile-verified
//
#include <hip/hip_runtime.h>
#include <math.h>

// ---------------- problem constants ----------------
#define Hh     4096
#define FW     12288            // W * 3 channels, flattened scalar columns
#define TILE   384              // output flattened columns per block (mult of 3, divides FW)
#define HALO   36               // 12-pixel radius * 3 channels
#define TOT    (TILE + 2*HALO)  // 456 loaded columns
#define BLOCK  480              // 15 wave32s (>= TOT, mult of 32)
#define SEG    512              // rows per segment
#define NSEG   (Hh / SEG)       // 8
#define NSTRIP (FW / TILE)      // 32
#define WARM   64               // IIR warm-up rows: decay^64 = e^-32 ~ 1.3e-14 << f32 eps
#define RAD    12               // blur radius (25-tap window)

typedef __attribute__((ext_vector_type(2))) float v2f;
typedef __attribute__((ext_vector_type(8))) float v8f;

// ---------------- CDNA5 async global->LDS path (probe-verified this toolchain) ----
#if defined(__gfx1250__) && \
    __has_builtin(__builtin_amdgcn_global_load_async_to_lds_b32) && \
    __has_builtin(__builtin_amdgcn_s_wait_asynccnt)
#define USE_ASYNC 1
#define AS1 __attribute__((address_space(1)))
#define AS3 __attribute__((address_space(3)))
#define ASYNC_CP_B32(gp, lp)                                                   \
  __builtin_amdgcn_global_load_async_to_lds_b32(                               \
      (AS1 int*)(const_cast<float*>(gp)), (AS3 int*)(lp), 0, 0)
#define ASYNC_WAIT(n) __builtin_amdgcn_s_wait_asynccnt(n)
#else
#define USE_ASYNC 0
#endif

// ---------------- CDNA5 WMMA path for the 3x3 channel-coupling GEMM ----------
#if defined(__gfx1250__) && __has_builtin(__builtin_amdgcn_wmma_f32_16x16x4_f32)
#define USE_WMMA 1
#else
#define USE_WMMA 0
#endif

__global__ __launch_bounds__(BLOCK) void chemdev_fused_kernel(
    const float* __restrict__ D,    // (H, W, 3) fp32
    const float* __restrict__ CM,   // (3, 3) fp32 coupling
    float* __restrict__ out)        // (H, W, 3) fp32
{
  __shared__ float stage[2][TOT];   // async-DMA row staging (double buffered)
  __shared__ float sv[TOT];         // vertical-blur row (with halo)
  __shared__ float si[TILE];        // inhibitor row (output cols only)
  __shared__ float so[TILE];        // coupled inhibition row (WMMA output)

  const int tid   = threadIdx.x;
  const int strip = blockIdx.x;
  const int seg   = blockIdx.y;
  const int base  = strip * TILE;   // base%3 == 0 -> pixels never straddle blocks
  const int r0    = seg * SEG;

  const int  cl      = base - HALO + tid;            // loaded flattened column
  const bool haveCol = (tid < TOT);
  const bool colOK   = haveCol && (cl >= 0) && (cl < FW);
  const bool isOut   = (tid >= HALO) && (tid < HALO + TILE);
  const int  otid    = isOut ? (tid - HALO) : 0;
  const int  fcol    = base + otid;
  const int  ch      = otid % 3;

  // Gaussian weights: kx = ky = exp(-2*d^2)/sum, d = -12..12 (sigma 0.5, ratio 1)
  float kw[RAD + 1];
  {
    kw[0] = 1.0f;
    float s = 1.0f;
#pragma unroll
    for (int d = 1; d <= RAD; ++d) {
      kw[d] = expf(-2.0f * (float)(d * d));
      s += 2.0f * kw[d];
    }
    const float inv = 1.0f / s;
#pragma unroll
    for (int d = 0; d <= RAD; ++d) kw[d] *= inv;
  }

  const float DEC  = 0.60653065971263342f;  // exp(-1/sigma_soft), < 0.999 clip
  const float OMD  = 1.0f - DEC;
  const float INV3 = 1.0f / 3.000001f;      // 1/(d_max - d_min + 1e-6) == 1/(supply+1e-6)

#if USE_WMMA
  // B (4x16 fp32): rows 0..2 = coupling matrix rows, rest zero.
  // Lane n(0-15): V0=B[0][n], V1=B[1][n]; lane n(16-31): V0=B[2][n-16], V1=B[3][..]=0.
  v2f Bm;
  {
    const int lane = tid & 31;
    const int nn = lane & 15;
    float b0 = 0.0f, b1 = 0.0f;
    if (nn < 3) {
      if (lane < 16) { b0 = CM[0 * 3 + nn]; b1 = CM[1 * 3 + nn]; }
      else           { b0 = CM[2 * 3 + nn]; }
    }
    Bm.x = b0; Bm.y = b1;
  }
#else
  const float C0 = CM[ch], C1 = CM[3 + ch], C2 = CM[6 + ch];
#endif

  // ---- IIR warm-up (exact for seg 0; decay^WARM negligible otherwise) ----
  float y = 0.0f;
  if (isOut) {
    int ws = r0 - WARM; if (ws < 0) ws = 0;
    for (int row = ws; row < r0; ++row)
      y = D[(size_t)row * FW + fcol] + DEC * y;
  }

  // ---- prefill 25-row register ring: ring[i] <-> row r0-13+i ----
  float ring[2 * RAD + 1];
#pragma unroll
  for (int i = 0; i < 2 * RAD + 1; ++i) {
    const int row = r0 - 13 + i;
    float v = 0.0f;
    if (colOK && row >= 0 && row < Hh) v = D[(size_t)row * FW + cl];
    ring[i] = v;
  }

  // row-issue helper: DMA row -> stage[row&1] (zero-fill out-of-bounds lanes)
  auto issue_row = [&](int row) {
    if (haveCol) {
      const bool ok = colOK && (row < Hh);
      float* sp = &stage[row & 1][tid];
#if USE_ASYNC
      if (ok) { ASYNC_CP_B32(&D[(size_t)row * FW + cl], sp); }
      else    { *sp = 0.0f; }
#else
      *sp = ok ? D[(size_t)row * FW + cl] : 0.0f;
#endif
    }
  };

  issue_row(r0 + 12);  // pre-issue so the loop runs one row ahead

  for (int r = r0; r < r0 + SEG; ++r) {
    issue_row(r + 13);
#if USE_ASYNC
    // row r+12 must be resident; only allow the just-issued row to stay in flight
    if (r + 13 < Hh) { ASYNC_WAIT(1); } else { ASYNC_WAIT(0); }
#endif
    __syncthreads();

    float x_r = 0.0f;
    if (haveCol) {
      const float xn = stage[(r + 12) & 1][tid];
#pragma unroll
      for (int i = 0; i < 2 * RAD; ++i) ring[i] = ring[i + 1];
      ring[2 * RAD] = xn;                 // ring now holds rows r-12 .. r+12
      float v = kw[0] * ring[RAD];
#pragma unroll
      for (int d = 1; d <= RAD; ++d) v += kw[d] * (ring[RAD - d] + ring[RAD + d]);
      sv[tid] = v;
      x_r = ring[RAD];                    // input row r (for IIR + mask)
    }
    __syncthreads();

    if (isOut) {
      y = x_r + DEC * y;
      const float soft = OMD * y;
      float hard = kw[0] * sv[tid];
#pragma unroll
      for (int d = 1; d <= RAD; ++d) hard += kw[d] * (sv[tid - 3 * d] + sv[tid + 3 * d]);
      float m = x_r * INV3;
      m = fminf(fmaxf(m, 0.0f), 1.0f);
      si[otid] = hard * m + soft * (1.0f - m);
    }
    __syncthreads();

#if USE_WMMA
    // Channel-coupling GEMM: (128 pixels x 3ch) * (3x3) via 8 wave-level
    // V_WMMA_F32_16X16X4_F32 ops (16 pixels per wave, K/N zero-padded to 4/16).
    {
      const int wk = tid >> 5;            // wave index (uniform per wave)
      if (wk < 8) {
        const int lane = tid & 31;
        const int p = 16 * wk + (lane & 15);   // pixel handled by this lane
        v2f Am;                            // A 16x4: lanes 0-15 K={0,1}, 16-31 K={2,3}
        Am.x = (lane < 16) ? si[3 * p]     : si[3 * p + 2];
        Am.y = (lane < 16) ? si[3 * p + 1] : 0.0f;
        v8f acc = {};
        v8f dm = __builtin_amdgcn_wmma_f32_16x16x4_f32(
            false, Am, false, Bm, (short)0, acc, false, false);
        if ((lane & 15) < 3) {             // only channel columns 0..2 are real
          const int half = (lane < 16) ? 0 : 8;
#pragma unroll
          for (int v = 0; v < 8; ++v)
            so[3 * (16 * wk + v + half) + (lane & 15)] = dm[v];
        }
      }
    }
    __syncthreads();
    if (isOut) {
      out[(size_t)r * FW + fcol] = 3.0f * tanhf((x_r - so[otid]) * INV3);
    }
#else
    if (isOut) {
      const int p0 = otid - ch;           // pixel base (all 3 channels in si)
      const float inhib = C0 * si[p0] + C1 * si[p0 + 1] + C2 * si[p0 + 2];
      out[(size_t)r * FW + fcol] = 3.0f * tanhf((x_r - inhib) * INV3);
    }
#endif
  }
}

extern "C" void kernel_launch(void* const* d_in, const int* in_sizes, int n_in,
                              void* d_out, int out_size, void* d_ws, size_t ws_size,
                              hipStream_t stream) {
  const float* D  = (const float*)d_in[0];
  const float* CM = (const float*)d_in[1];
  float* out = (float*)d_out;
  dim3 grid(NSTRIP, NSEG);
  chemdev_fused_kernel<<<grid, BLOCK, 0, stream>>>(D, CM, out);
}